// TaylorSoftmax_12429635354923
// MI455X (gfx1250) — compile-verified
//
#include <hip/hip_runtime.h>
#include <hip/hip_bf16.h>
#include <stdint.h>

#ifndef __has_builtin
#define __has_builtin(x) 0
#endif

#define TS_EPS 1e-8f

// Native clang vector types (swizzle-capable, work with nontemporal builtins)
typedef float         f32x4 __attribute__((ext_vector_type(4)));
typedef unsigned int  u32x4 __attribute__((ext_vector_type(4)));
typedef int           i32x8 __attribute__((ext_vector_type(8)));
typedef int           i32x4 __attribute__((ext_vector_type(4)));

#if defined(__gfx1250__) && __has_builtin(__builtin_amdgcn_tensor_load_to_lds) && \
    __has_builtin(__builtin_amdgcn_s_wait_tensorcnt)
#define TS_USE_TDM 1
#else
#define TS_USE_TDM 0
#endif

__device__ __forceinline__ float waveMax32(float v) {
#pragma unroll
    for (int off = 16; off > 0; off >>= 1)
        v = fmaxf(v, __shfl_xor(v, off, 32));
    return v;
}

__device__ __forceinline__ float waveSum32(float v) {
#pragma unroll
    for (int off = 16; off > 0; off >>= 1)
        v += __shfl_xor(v, off, 32);
    return v;
}

// One workgroup per row. Row (C fp32, C%4==0) is staged in LDS once via the
// Tensor Data Mover, processed entirely from LDS, and streamed out with
// non-temporal b128 stores.  Scratch layout: smem[0..C) row, smem[C..C+32)
// max partials, smem[C+32..C+64) sum partials.
extern "C" __global__ __launch_bounds__(512)
void taylor_softmax_row_kernel(const float* __restrict__ logits,
                               float* __restrict__ out,
                               int C)
{
    extern __shared__ float smem[];
    float* rowLds  = smem;
    float* maxScr  = smem + C;
    float* sumScr  = smem + C + 32;

    const int tid   = threadIdx.x;
    const int nthr  = blockDim.x;
    const int lane  = tid & 31;
    const int wave  = tid >> 5;
    const int nwave = nthr >> 5;
    const int C4    = C >> 2;

    const long long rbase = (long long)blockIdx.x * (long long)C;
    const float* __restrict__ grow = logits + rbase;
    float* __restrict__       gout = out + rbase;

#if TS_USE_TDM
    if (wave == 0) {
        // ---- Tensor DMA descriptor (D#), 1-D tile: C fp32 elements ----
        const uint64_t ga       = (uint64_t)(uintptr_t)grow;
        const unsigned lds_base = __builtin_amdgcn_groupstaticsize(); // dyn LDS start

        u32x4 g0;
        g0.x = 1u;                                   // count=1 (valid user D#)
        g0.y = lds_base;                             // lds_addr (bytes)
        g0.z = (unsigned)(ga & 0xFFFFFFFFu);         // global_addr[31:0]
        g0.w = (unsigned)((ga >> 32) & 0x01FFFFFFu)  // global_addr[56:32]
             | (2u << 30);                           // type=2 ("image")

        i32x8 g1;
        g1[0] = (int)(2u << 16);                             // data_size=4B; wg_mask=0
        g1[1] = (int)(((unsigned)C & 0xFFFFu) << 16);        // tensor_dim0[15:0]
        g1[2] = (int)(((unsigned)C >> 16) | (1u << 16));     // tensor_dim0[31:16] | tensor_dim1=1
        g1[3] = (int)(((unsigned)C & 0xFFFFu) << 16);        // tile_dim0 = C
        g1[4] = 1;                                           // tile_dim1 = 1, tile_dim2 = 0
        g1[5] = C;                                           // tensor_dim0_stride[31:0]
        g1[6] = (int)(((unsigned)C & 0xFFFFu) << 16);        // dim0_stride hi=0 | dim1_stride lo
        g1[7] = (int)((unsigned)C >> 16);                    // dim1_stride hi

        i32x4 gz4 = {0, 0, 0, 0};                            // groups 2/3 unused (<=2D)
        i32x8 gz8 = {0, 0, 0, 0, 0, 0, 0, 0};                // extra group word (clang-23 form)

        // amdgpu-toolchain (clang-23) 6-arg form.
        __builtin_amdgcn_tensor_load_to_lds(g0, g1, gz4, gz4, gz8, 0);
        __builtin_amdgcn_s_wait_tensorcnt(0);
    }
    __syncthreads();   // LDS row visible to all waves
#else
    // Fallback: cooperative b128 global->LDS staging (single HBM read).
    for (int i = tid; i < C4; i += nthr) {
        f32x4 v = __builtin_nontemporal_load((const f32x4*)grow + i);
        ((f32x4*)rowLds)[i] = v;
    }
    __syncthreads();
#endif

    // ---- Pass 1: row max (LDS b128 reads) ----
    float m = -__FLT_MAX__;
    for (int i = tid; i < C4; i += nthr) {
        f32x4 v = ((const f32x4*)rowLds)[i];
        m = fmaxf(m, fmaxf(fmaxf(v.x, v.y), fmaxf(v.z, v.w)));
    }
    m = waveMax32(m);
    if (lane == 0) maxScr[wave] = m;
    __syncthreads();
    if (wave == 0) {
        float t = (lane < nwave) ? maxScr[lane] : -__FLT_MAX__;
        t = waveMax32(t);
        if (lane == 0) maxScr[0] = t;
    }
    __syncthreads();
    m = maxScr[0];

    // ---- Pass 2: numer = 1 + x + (x+eps)^2/2, accumulate sum, overwrite LDS ----
    float s = 0.0f;
    for (int i = tid; i < C4; i += nthr) {
        f32x4 v = ((const f32x4*)rowLds)[i];
        f32x4 n;
        {
            float x = v.x - m, t = x + TS_EPS;
            n.x = fmaf(0.5f * t, t, 1.0f + x);
        }
        {
            float x = v.y - m, t = x + TS_EPS;
            n.y = fmaf(0.5f * t, t, 1.0f + x);
        }
        {
            float x = v.z - m, t = x + TS_EPS;
            n.z = fmaf(0.5f * t, t, 1.0f + x);
        }
        {
            float x = v.w - m, t = x + TS_EPS;
            n.w = fmaf(0.5f * t, t, 1.0f + x);
        }
        s += (n.x + n.y) + (n.z + n.w);
        ((f32x4*)rowLds)[i] = n;
    }
    s = waveSum32(s);
    if (lane == 0) sumScr[wave] = s;
    __syncthreads();
    if (wave == 0) {
        float t = (lane < nwave) ? sumScr[lane] : 0.0f;
        t = waveSum32(t);
        if (lane == 0) sumScr[0] = t;
    }
    __syncthreads();
    const float inv = 1.0f / (sumScr[0] + TS_EPS);

    // ---- Pass 3: normalize + streaming non-temporal b128 stores ----
    for (int i = tid; i < C4; i += nthr) {
        f32x4 v = ((const f32x4*)rowLds)[i];
        v.x *= inv; v.y *= inv; v.z *= inv; v.w *= inv;
        __builtin_nontemporal_store(v, (f32x4*)gout + i);
    }
}

extern "C" void kernel_launch(void* const* d_in, const int* in_sizes, int n_in,
                              void* d_out, int out_size, void* d_ws, size_t ws_size,
                              hipStream_t stream) {
    (void)n_in; (void)d_ws; (void)ws_size; (void)out_size;

    const float* logits = (const float*)d_in[0];
    float* out          = (float*)d_out;

    const int C = 32000;                 // row length (vocab), matches reference
    const int B = in_sizes[0] / C;       // 8192 rows

    // Row (C fp32) + 64 floats of reduction scratch in dynamic LDS:
    // 128,256 B < 320 KB WGP LDS; two blocks co-resident per WGP.
    const size_t shmem = (size_t)(C + 64) * sizeof(float);

    // Allow >64KB dynamic LDS (MI455X WGP has 320KB).
    (void)hipFuncSetAttribute(reinterpret_cast<const void*>(taylor_softmax_row_kernel),
                              hipFuncAttributeMaxDynamicSharedMemorySize,
                              (int)shmem);

    taylor_softmax_row_kernel<<<B, 512, shmem, stream>>>(logits, out, C);
}